// BuildCost_23192823399035
// MI455X (gfx1250) — compile-verified
//
#include <hip/hip_runtime.h>

typedef float v2f __attribute__((ext_vector_type(2)));
typedef float v8f __attribute__((ext_vector_type(8)));

#define AVIEWS 9
#define AA     81         // A*A taps
#define AAP    84         // taps padded to K multiple of 4
#define C0V    4          // A/2
#define Hc     192
#define Wc     192
#define HWc    (Hc * Wc)
#define NCH    32         // groups (c)
#define MPG    4          // outputs per group (cout/c)
#define ND     9          // disparities (-4..4)
#define KCH    21         // 84/4 K-chunks
#define TILE   16         // pixels per WMMA M tile (contiguous in a row)
#define TPR    (Wc / TILE)          // 12 tiles per image row
#define NTILES (Hc * TPR)           // 2304
#define WPB    8                    // waves per block

// pair-swizzled LDS index: (tap,pix) -> ((tap/2)*16 + pix)*2 + (tap&1)
// so taps {2t, 2t+1} for one lane are adjacent -> single ds_load_b64
__device__ __forceinline__ int pswz(int tap, int pix) {
    return (((tap >> 1) * TILE + pix) << 1) + (tap & 1);
}
__device__ __forceinline__ int clamp191(int v) { return min(max(v, 0), Wc - 1); }

template <bool BORDER>
__device__ __forceinline__ void run_k(const float* __restrict__ x,
                                      const float* __restrict__ smaskw,
                                      const float* __restrict__ sWbc,
                                      int c, int i, int jl, int half, int l16,
                                      v8f acc[ND])
{
    for (int k0 = 0; k0 < KCH; ++k0) {
        const int tap0  = k0 * 4 + half * 2;         // even; A lanes 0-15: K={4k,4k+1}; 16-31: {4k+2,4k+3}
        const int tapc0 = min(tap0, AA - 1);         // clamp for safe addressing (mask==0 kills pad taps)
        const int tapc1 = min(tap0 + 1, AA - 1);

        // B pair + mask pair: one ds_load_b64 each (pair-swizzled layouts)
        const v2f bm = *(const v2f*)&sWbc  [pswz(tap0, l16)];
        const v2f mp = *(const v2f*)&smaskw[pswz(tap0, l16)];
        const float m0 = mp.x, m1 = mp.y;            // 0 for pad taps

        const int p0 = tapc0 / AVIEWS, q0 = tapc0 - p0 * AVIEWS;
        const int p1 = tapc1 / AVIEWS, q1 = tapc1 - p1 * AVIEWS;
        const int dr0 = C0V - p0, dq0 = C0V - q0;
        const int dr1 = C0V - p1, dq1 = C0V - q1;
        const unsigned base0 = (unsigned)(c * AA + tapc0) * (unsigned)HWc;
        const unsigned base1 = (unsigned)(c * AA + tapc1) * (unsigned)HWc;

        float xv0[ND], xv1[ND], mv0[ND], mv1[ND];

        // ---- phase 1: issue all 18 gathers (32-bit offsets -> saddr+voffset scale_offset) ----
        #pragma unroll
        for (int dIdx = 0; dIdx < ND; ++dIdx) {
            const int d = dIdx - 4;
            int r0 = i + dr0 * d, c0 = jl + dq0 * d;
            int r1 = i + dr1 * d, c1 = jl + dq1 * d;
            if (BORDER) {
                bool in0 = ((unsigned)r0 < (unsigned)Hc) && ((unsigned)c0 < (unsigned)Wc);
                bool in1 = ((unsigned)r1 < (unsigned)Hc) && ((unsigned)c1 < (unsigned)Wc);
                xv0[dIdx] = x[base0 + (unsigned)(clamp191(r0) * Wc + clamp191(c0))];
                xv1[dIdx] = x[base1 + (unsigned)(clamp191(r1) * Wc + clamp191(c1))];
                mv0[dIdx] = in0 ? m0 : 0.f;          // v_cndmask; independent of loads
                mv1[dIdx] = in1 ? m1 : 0.f;
            } else {                                  // interior: shifts (<=16px) provably in-bounds
                xv0[dIdx] = x[base0 + (unsigned)(r0 * Wc + c0)];
                xv1[dIdx] = x[base1 + (unsigned)(r1 * Wc + c1)];
                mv0[dIdx] = m0;
                mv1[dIdx] = m1;
            }
        }

        // ---- phase 2: consume + 9 WMMAs ----
        #pragma unroll
        for (int dIdx = 0; dIdx < ND; ++dIdx) {
            v2f am;
            am.x = xv0[dIdx] * mv0[dIdx];
            am.y = xv1[dIdx] * mv1[dIdx];
            acc[dIdx] = __builtin_amdgcn_wmma_f32_16x16x4_f32(
                false, am, false, bm, (short)0, acc[dIdx], false, false);
        }
    }
}

__global__ __launch_bounds__(256) void buildcost_wmma_f32(
    const float* __restrict__ x,     // [32][81][192][192]
    const float* __restrict__ mask,  // [81][192][192]
    const float* __restrict__ Wt,    // [128][81]
    float* __restrict__ out)         // [128][9][192][192]
{
    __shared__ float smask[WPB][AAP * TILE];       // mask tile per wave (pair-swizzled), taps>=81 zeroed
    __shared__ float sinv [WPB][TILE];             // 81 / sum(mask) per pixel
    __shared__ float sout [WPB][ND * MPG * TILE];  // output staging per wave
    __shared__ float sWb  [4][AAP * TILE];         // B operand (pair-swizzled), block-shared

    const int lane = threadIdx.x & 31;
    const int wave = threadIdx.x >> 5;
    const int tile = blockIdx.x * WPB + wave;      // adjacent waves -> adjacent tiles (L2 shear reuse)
    const int i    = tile / TPR;
    const int j0   = (tile % TPR) * TILE;
    const int cGrp = blockIdx.y;                   // 0..7, channel slice

    // ---- block-cooperative B table: sWb[cs][pswz(tap,n)] = (n<4 && tap<81) ? W[(c*4+n)*81+tap] : 0
    for (int idx = threadIdx.x; idx < 4 * AAP * TILE; idx += 256) {
        int cs  = idx / (AAP * TILE);
        int rem = idx - cs * (AAP * TILE);
        int tap = rem >> 4;
        int n   = rem & 15;
        float v = 0.f;
        if (n < MPG && tap < AA)
            v = Wt[(((cGrp + 8 * cs) * MPG + n) * AA) + tap];
        sWb[cs][pswz(tap, n)] = v;
    }

    // ---- per-wave mask tile (pair-swizzled), taps >= 81 zeroed ----
    for (int idx = lane; idx < AAP * TILE; idx += 32) {
        int tap = idx >> 4;
        int pix = idx & 15;
        smask[wave][pswz(tap, pix)] =
            (tap < AA) ? mask[tap * HWc + i * Wc + j0 + pix] : 0.f;
    }
    __syncthreads();
    if (lane < TILE) {
        float s = 0.f;
        for (int tap = 0; tap < AA; ++tap) s += smask[wave][pswz(tap, lane)];
        sinv[wave][lane] = (float)AA / s;           // divide-by-mean == *81/sum
    }
    __syncthreads();

    const int half = lane >> 4;
    const int l16  = lane & 15;
    const int jl   = j0 + l16;
    // wave-uniform: are all sheared reads (|shift|<=16) in-bounds for this tile?
    const bool border = (i < 2 * TILE) || (i >= Hc - 2 * TILE) ||
                        (j0 < TILE)    || (j0 + TILE > Wc - TILE);

    for (int cs = 0; cs < 4; ++cs) {
        const int c = cGrp + 8 * cs;
        v8f acc[ND] = {};

        if (border) run_k<true >(x, smask[wave], sWb[cs], c, i, jl, half, l16, acc);
        else        run_k<false>(x, smask[wave], sWb[cs], c, i, jl, half, l16, acc);

        // ---- stage D -> LDS (barrier guards WAR vs previous iteration's reads) ----
        __syncthreads();
        if (l16 < MPG) {
            #pragma unroll
            for (int dIdx = 0; dIdx < ND; ++dIdx) {
                #pragma unroll
                for (int r = 0; r < 8; ++r) {
                    int pix = r + 8 * half;        // C/D: VGPR r -> M = r + 8*half
                    sout[wave][(dIdx * MPG + l16) * TILE + pix] = acc[dIdx][r];
                }
            }
        }
        __syncthreads();

        // ---- coalesced scaled store: 16-consecutive-pixel runs per (o,d) ----
        for (int idx = lane; idx < ND * MPG * TILE; idx += 32) {
            int pix  = idx & 15;
            int dm   = idx >> 4;
            int m    = dm & 3;
            int dIdx = dm >> 2;
            int o    = c * MPG + m;
            out[(((size_t)o * ND + dIdx) * Hc + i) * Wc + j0 + pix] =
                sout[wave][idx] * sinv[wave][pix];
        }
    }
}

extern "C" void kernel_launch(void* const* d_in, const int* in_sizes, int n_in,
                              void* d_out, int out_size, void* d_ws, size_t ws_size,
                              hipStream_t stream) {
    const float* x    = (const float*)d_in[0];   // [1,32,81,192,192]
    const float* mask = (const float*)d_in[1];   // [1,81,192,192]
    const float* Wt   = (const float*)d_in[2];   // [128,81]
    float* out        = (float*)d_out;           // [1,128,9,192,192]

    dim3 grid(NTILES / WPB, 8);                  // 288 x 8 blocks
    dim3 block(32 * WPB);                        // 8 wave32 per block
    buildcost_wmma_f32<<<grid, block, 0, stream>>>(x, mask, Wt, out);
}